// AggregationLayer_317827580221
// MI455X (gfx1250) — compile-verified
//
#include <hip/hip_runtime.h>

// MI455X (gfx1250) — bandwidth-bound masked segmented reduction recast as
// one-hot GEMM on the WMMA pipe (V_WMMA_F32_16X16X4_F32).
// Traffic: 16*480*640 px * 44 B = ~216 MB @ 23.3 TB/s -> ~9.3 us floor.
// Round 2: 4 pixels/lane, b128 global loads + b128 LDS staging to cut
// dynamic instruction count ~35% (issue rate was near the HBM floor).

typedef __attribute__((ext_vector_type(2))) float v2f;
typedef __attribute__((ext_vector_type(4))) float v4f;
typedef __attribute__((ext_vector_type(8))) float v8f;
typedef __attribute__((ext_vector_type(4))) int   v4i;

#define B_SZ   16
#define H_SZ   480
#define W_SZ   640
#define HW     (H_SZ * W_SZ)      // 307200
#define NBLK   75                 // blocks per batch image
#define PPB    (HW / NBLK)        // 4096 pixels per block
#define CHUNK  1024               // pixels per block-iteration (8 waves * 128)
#define ITERS  (PPB / CHUNK)      // 4
#define ROWP   132                // LDS row stride (dwords): 16B-aligned b128 stores,
                                  // conflict-free transpose gather (banks 4m+p0)
#define EPSF   1e-6f

// ---------------------------------------------------------------- zero ws
__global__ void zero_ws_kernel(float* __restrict__ ws, int n) {
    int i = blockIdx.x * blockDim.x + threadIdx.x;
    if (i < n) ws[i] = 0.0f;
}

// ---------------------------------------------------------------- main reduce
// Channels (N of the WMMA tile): 0-3 quat, 4-6 scales, 7 z, 8 Mxx, 9 Mxy,
// 10 Myy, 11 rx, 12 ry, 13 ones (mask size). Rows (M): class id 0..15.
__global__ __launch_bounds__(256) void agg_reduce_kernel(
    const int*   __restrict__ cat,
    const float* __restrict__ quat,
    const float* __restrict__ scl,
    const float* __restrict__ xy,
    const float* __restrict__ zc,
    float*       __restrict__ ws)
{
    // Per-wave staging: rows 0..13 channels, row 14 cat bits; 128 px per row.
    __shared__ float feat[8][15 * ROWP];          // ~62 KB
    float* red = &feat[0][0];                     // aliased after barrier

    const int tid  = threadIdx.x;
    const int lane = tid & 31;
    const int wave = tid >> 5;
    const int b    = blockIdx.y;
    const int pbase = blockIdx.x * PPB;

    float* wlds = feat[wave];
    const int m    = lane & 15;   // class row for A, channel col for B
    const int half = lane >> 4;

    const int catBase = b * HW;
    const int qBase   = b * 4 * HW;
    const int sBase   = b * 3 * HW;
    const int xyBase  = b * 2 * HW;
    const int zBase   = b * HW;

    v8f acc = {};   // 16x16 f32 accumulator tile (classes x channels)

    for (int it = 0; it < ITERS; ++it) {
        // This lane owns 4 consecutive pixels (never crossing a row: 640%4==0).
        const int pB = pbase + it * CHUNK + wave * 128 + 4 * lane;
        const int pw0 = pB % W_SZ;
        const int ph0 = pB / W_SZ;

        // Vectorized coalesced streaming loads (stream > 192 MB L2 -> NT).
        const v4i cm  = __builtin_nontemporal_load((const v4i*)(cat  + catBase + pB));
        const v4f q0v = __builtin_nontemporal_load((const v4f*)(quat + qBase + 0 * HW + pB));
        const v4f q1v = __builtin_nontemporal_load((const v4f*)(quat + qBase + 1 * HW + pB));
        const v4f q2v = __builtin_nontemporal_load((const v4f*)(quat + qBase + 2 * HW + pB));
        const v4f q3v = __builtin_nontemporal_load((const v4f*)(quat + qBase + 3 * HW + pB));
        const v4f s0v = __builtin_nontemporal_load((const v4f*)(scl  + sBase + 0 * HW + pB));
        const v4f s1v = __builtin_nontemporal_load((const v4f*)(scl  + sBase + 1 * HW + pB));
        const v4f s2v = __builtin_nontemporal_load((const v4f*)(scl  + sBase + 2 * HW + pB));
        const v4f xv  = __builtin_nontemporal_load((const v4f*)(xy   + xyBase + 0 * HW + pB));
        const v4f yv  = __builtin_nontemporal_load((const v4f*)(xy   + xyBase + 1 * HW + pB));
        const v4f zv  = __builtin_nontemporal_load((const v4f*)(zc   + zBase + pB));

        v4f Mxxv, Mxyv, Myyv, rxv, ryv;
        const float fh = (float)ph0;
#pragma unroll
        for (int k = 0; k < 4; ++k) {
            const float x   = xv[k];
            const float y   = yv[k];
            const float nrm = sqrtf(x * x + y * y) + EPSF;
            const float dx  = x / nrm;
            const float dy  = y / nrm;
            const float Mxx = 1.0f - dx * dx;
            const float Mxy = -dx * dy;
            const float Myy = 1.0f - dy * dy;
            const float fw  = (float)(pw0 + k);
            Mxxv[k] = Mxx;
            Mxyv[k] = Mxy;
            Myyv[k] = Myy;
            rxv[k]  = Mxx * fw + Mxy * fh;
            ryv[k]  = Mxy * fw + Myy * fh;
        }

        // Channel-major b128 LDS staging (wave-private; per-wave DS is in-order).
        const int col = 4 * lane;
        *(v4f*)&wlds[0  * ROWP + col] = q0v;
        *(v4f*)&wlds[1  * ROWP + col] = q1v;
        *(v4f*)&wlds[2  * ROWP + col] = q2v;
        *(v4f*)&wlds[3  * ROWP + col] = q3v;
        *(v4f*)&wlds[4  * ROWP + col] = s0v;
        *(v4f*)&wlds[5  * ROWP + col] = s1v;
        *(v4f*)&wlds[6  * ROWP + col] = s2v;
        *(v4f*)&wlds[7  * ROWP + col] = zv;
        *(v4f*)&wlds[8  * ROWP + col] = Mxxv;
        *(v4f*)&wlds[9  * ROWP + col] = Mxyv;
        *(v4f*)&wlds[10 * ROWP + col] = Myyv;
        *(v4f*)&wlds[11 * ROWP + col] = rxv;
        *(v4f*)&wlds[12 * ROWP + col] = ryv;
        *(v4f*)&wlds[13 * ROWP + col] = (v4f){1.0f, 1.0f, 1.0f, 1.0f};
        *(v4i*)&wlds[14 * ROWP + col] = cm;

        // 32 WMMAs consume the 128-pixel chunk: A = one-hot(class), B = channels^T.
        // f32 16x16x4 layouts: VGPR j, lane-half h -> K = j + 2*h (A and B alike).
#pragma unroll
        for (int g = 0; g < 32; ++g) {
            const int p0 = g * 4 + 2 * half;
            const int c0 = __float_as_int(wlds[14 * ROWP + p0]);     // broadcast read
            const int c1 = __float_as_int(wlds[14 * ROWP + p0 + 1]); // broadcast read
            v2f a;
            a.x = (c0 == m) ? 1.0f : 0.0f;
            a.y = (c1 == m) ? 1.0f : 0.0f;
            v2f bb;
            bb.x = wlds[m * ROWP + p0];       // banks 4m+p0: conflict-free gather
            bb.y = wlds[m * ROWP + p0 + 1];
            acc = __builtin_amdgcn_wmma_f32_16x16x4_f32(
                false, a, false, bb, (short)0, acc, false, false);
        }
    }

    // Cross-wave reduction of the 16x16 tiles (red aliases feat -> barrier first).
    __syncthreads();
#pragma unroll
    for (int v = 0; v < 8; ++v) red[wave * 256 + v * 32 + lane] = acc[v];
    __syncthreads();

    float s = 0.0f;
#pragma unroll
    for (int w2 = 0; w2 < 8; ++w2) s += red[w2 * 256 + tid];

    const int v  = tid >> 5;
    const int L  = tid & 31;
    const int mm = v + 8 * (L >> 4);   // C/D layout: VGPR v, lane-half h -> M = v + 8h
    const int nn = L & 15;
    atomicAdd(&ws[b * 256 + mm * 16 + nn], s);
}

// ---------------------------------------------------------------- epilogue
__global__ void agg_finish_kernel(const float* __restrict__ ws,
                                  const float* __restrict__ K,
                                  float*       __restrict__ out)
{
    const int t = threadIdx.x;
    if (t >= 96) return;
    const int b = t & 15;
    const int c = 1 + (t >> 4);   // classes 1..6

    float S[14];
#pragma unroll
    for (int n = 0; n < 14; ++n) S[n] = ws[b * 256 + c * 16 + n];

    const float denom = fmaxf(S[13], 1.0f);
    const float invd  = 1.0f / denom;
    const float q0 = S[0] * invd, q1 = S[1] * invd, q2 = S[2] * invd, q3 = S[3] * invd;
    const float s0 = S[4] * invd, s1 = S[5] * invd, s2 = S[6] * invd;
    const float zp = S[7] * invd;

    const float Axx = S[8] + EPSF, Axy = S[9], Ayy = S[10] + EPSF;
    const float rx = S[11], ry = S[12];
    const float det = Axx * Ayy - Axy * Axy;
    const float cx  = (Ayy * rx - Axy * ry) / det;
    const float cy  = (Axx * ry - Axy * rx) / det;

    const float qn = sqrtf(q0 * q0 + q1 * q1 + q2 * q2 + q3 * q3) + 1e-8f;
    const float w = q0 / qn, x = q1 / qn, y = q2 / qn, z = q3 / qn;
    const float R00 = 1.0f - 2.0f * (y * y + z * z), R01 = 2.0f * (x * y - w * z), R02 = 2.0f * (x * z + w * y);
    const float R10 = 2.0f * (x * y + w * z), R11 = 1.0f - 2.0f * (x * x + z * z), R12 = 2.0f * (y * z - w * x);
    const float R20 = 2.0f * (x * z - w * y), R21 = 2.0f * (y * z + w * x), R22 = 1.0f - 2.0f * (x * x + y * y);

    const float ez = expf(zp);

    // general 3x3 inverse of intrinsics
    const float k0 = K[0], k1 = K[1], k2 = K[2], k3 = K[3], k4 = K[4],
                k5 = K[5], k6 = K[6], k7 = K[7], k8 = K[8];
    const float d3  = k0 * (k4 * k8 - k5 * k7) - k1 * (k3 * k8 - k5 * k6) + k2 * (k3 * k7 - k4 * k6);
    const float id3 = 1.0f / d3;
    const float i00 = (k4 * k8 - k5 * k7) * id3, i01 = (k2 * k7 - k1 * k8) * id3, i02 = (k1 * k5 - k2 * k4) * id3;
    const float i10 = (k5 * k6 - k3 * k8) * id3, i11 = (k0 * k8 - k2 * k6) * id3, i12 = (k2 * k3 - k0 * k5) * id3;
    const float i20 = (k3 * k7 - k4 * k6) * id3, i21 = (k1 * k6 - k0 * k7) * id3, i22 = (k0 * k4 - k1 * k3) * id3;

    const float t0 = ez * (i00 * cx + i01 * cy + i02);
    const float t1 = ez * (i10 * cx + i11 * cy + i12);
    const float t2 = ez * (i20 * cx + i21 * cy + i22);

    float* o = out + ((c - 1) * B_SZ + b) * 26;
    o[0]  = q0;  o[1]  = q1;  o[2]  = q2;  o[3]  = q3;
    o[4]  = s0;  o[5]  = s1;  o[6]  = s2;  o[7]  = zp;
    o[8]  = cx;  o[9]  = cy;
    o[10] = R00; o[11] = R01; o[12] = R02; o[13] = t0;
    o[14] = R10; o[15] = R11; o[16] = R12; o[17] = t1;
    o[18] = R20; o[19] = R21; o[20] = R22; o[21] = t2;
    o[22] = 0.0f; o[23] = 0.0f; o[24] = 0.0f; o[25] = 1.0f;
}

// ---------------------------------------------------------------- launcher
extern "C" void kernel_launch(void* const* d_in, const int* in_sizes, int n_in,
                              void* d_out, int out_size, void* d_ws, size_t ws_size,
                              hipStream_t stream) {
    const int*   cat  = (const int*)d_in[0];
    const float* quat = (const float*)d_in[1];
    const float* scl  = (const float*)d_in[2];
    const float* xy   = (const float*)d_in[3];
    const float* zc   = (const float*)d_in[4];
    const float* K    = (const float*)d_in[5];
    float* out = (float*)d_out;
    float* ws  = (float*)d_ws;   // 16 batches * 256-float tile = 16 KB

    zero_ws_kernel<<<16, 256, 0, stream>>>(ws, B_SZ * 256);
    dim3 grid(NBLK, B_SZ);
    agg_reduce_kernel<<<grid, 256, 0, stream>>>(cat, quat, scl, xy, zc, ws);
    agg_finish_kernel<<<1, 128, 0, stream>>>(ws, K, out);
}